// ULossTopkPre_26697516712401
// MI455X (gfx1250) — compile-verified
//
#include <hip/hip_runtime.h>
#include <hip/hip_bf16.h>

typedef __attribute__((ext_vector_type(2))) float v2f;
typedef __attribute__((ext_vector_type(8))) float v8f;

#define Hh 512
#define Ww 512
#define HC 496              // 512 - 2*8
#define WC 496
#define NVV 49
#define SXH (Ww*3)          // 1536: h-stride of x[b][u][v][h][w][c]
#define SXN (Hh*Ww*3)       // 786432: per-view stride (n = u*7+v)
#define SXB (NVV*SXN)       // per-batch stride
#define CBASE (24*SXN)      // center view offset

__device__ __forceinline__ int dev_K(int epoch) {
    int k;
    if (epoch < 200)        k = 0;
    else if (epoch < 2300)  k = ((epoch - 200) / 100) * 2;
    else                    k = 44;
    return NVV - k;
}

// One thread per cropped pixel. Phase 1: 49 bilinear float3 samples -> |warped-center|
// staged in LDS (stride 147 floats/thread: odd -> bank-conflict-free on 64 banks).
// Phase 2: per channel, 49 values into registers, K-th-order-statistic threshold by
// rank counting (inner loop fully unrolled so v[] stays in VGPRs), masked sum.
__global__ __launch_bounds__(64) void color_kernel(
    const float* __restrict__ pred, const float* __restrict__ x,
    const int* __restrict__ epoch_p, float* __restrict__ partials)
{
    __shared__ float lds[64 * 147];
    const int tid = threadIdx.x;
    const int pid = blockIdx.x * 64 + tid;
    const int b   = pid / (HC * WC);
    const int rem = pid - b * (HC * WC);
    const int yy  = 8 + rem / WC;
    const int xx  = 8 + (rem - (rem / WC) * WC);

    const float p  = pred[(b * Hh + yy) * Ww + xx];
    const int   cb = b * SXB + CBASE + yy * SXH + xx * 3;
    float cen[3] = { x[cb], x[cb + 1], x[cb + 2] };

    float* myv = &lds[tid * 147];

    for (int n = 0; n < NVV; ++n) {
        const int un = n / 7, vn = n - un * 7;
        float ys = (float)yy + p * (float)(un - 3);
        float xs = (float)xx + p * (float)(vn - 3);
        ys = fminf(fmaxf(ys, 0.0f), 511.0f);
        xs = fminf(fmaxf(xs, 0.0f), 511.0f);
        const float fy = floorf(ys), fx = floorf(xs);
        const int y0 = (int)fy, x0 = (int)fx;
        const int y1 = min(y0 + 1, 511), x1 = min(x0 + 1, 511);
        const float wy = ys - fy, wx = xs - fx;
        const float w00 = (1.0f - wy) * (1.0f - wx);
        const float w01 = (1.0f - wy) * wx;
        const float w10 = wy * (1.0f - wx);
        const float w11 = wy * wx;
        const int bn = b * SXB + n * SXN;
        const float* p00 = x + bn + y0 * SXH + x0 * 3;
        const float* p01 = x + bn + y0 * SXH + x1 * 3;
        const float* p10 = x + bn + y1 * SXH + x0 * 3;
        const float* p11 = x + bn + y1 * SXH + x1 * 3;
        #pragma unroll
        for (int c = 0; c < 3; ++c) {
            float r = p00[c] * w00 + p01[c] * w01 + p10[c] * w10 + p11[c] * w11;
            myv[c * 49 + n] = fabsf(r - cen[c]);
        }
    }

    const int K = dev_K(epoch_p[0]);
    float acc = 0.0f;
    #pragma unroll 1
    for (int c = 0; c < 3; ++c) {
        const float* base = myv + c * 49;
        float v[49];
        #pragma unroll
        for (int n = 0; n < 49; ++n) v[n] = base[n];
        // threshold = smallest value whose rank-count (#<=) reaches K  == sorted[K-1]
        float thresh = 3.4e38f;
        #pragma unroll 1
        for (int i = 0; i < 49; ++i) {
            const float vi = base[i];
            int cnt = 0;
            #pragma unroll
            for (int j = 0; j < 49; ++j) cnt += (v[j] <= vi) ? 1 : 0;
            if (cnt >= K && vi < thresh) thresh = vi;
        }
        float s = 0.0f;
        #pragma unroll
        for (int j = 0; j < 49; ++j) s += (v[j] <= thresh) ? v[j] : 0.0f;
        acc += s;
    }

    __syncthreads();                 // done with per-thread LDS regions
    lds[tid] = acc;                  // reuse LDS for deterministic tree reduction
    __syncthreads();
    #pragma unroll
    for (int off = 32; off > 0; off >>= 1) {
        if (tid < off) lds[tid] += lds[tid + off];
        __syncthreads();
    }
    if (tid == 0) partials[blockIdx.x] = lds[0];
}

// Edge-aware smoothness: one thread per (b,y,x) term of lx then ly, block-reduced.
__global__ __launch_bounds__(256) void grad_kernel(
    const float* __restrict__ pred, const float* __restrict__ x,
    float* __restrict__ partials)
{
    const int NLX = 2 * 496 * 495;   // y in [8,504), xi in [8,503)
    const int NLY = 2 * 495 * 496;   // y in [8,503), xi in [8,504)
    const int gid = blockIdx.x * 256 + threadIdx.x;
    float val = 0.0f;
    if (gid < NLX) {
        const int b = gid / (496 * 495);
        const int r = gid - b * (496 * 495);
        const int y  = 8 + r / 495;
        const int xi = 8 + (r - (r / 495) * 495);
        const int cbp = b * SXB + CBASE + y * SXH + xi * 3;
        const float g = fabsf(x[cbp + 3 + 0] - x[cbp + 0])
                      + fabsf(x[cbp + 3 + 1] - x[cbp + 1])
                      + fabsf(x[cbp + 3 + 2] - x[cbp + 2]);
        const float w = expf(-50.0f * g);           // -150 * mean over 3 channels
        const int pb = (b * Hh + y) * Ww + xi;
        val = w * fabsf(pred[pb + 1] - pred[pb]);
    } else if (gid < NLX + NLY) {
        const int t = gid - NLX;
        const int b = t / (495 * 496);
        const int r = t - b * (495 * 496);
        const int y  = 8 + r / 496;
        const int xi = 8 + (r - (r / 496) * 496);
        const int cbp = b * SXB + CBASE + y * SXH + xi * 3;
        const float g = fabsf(x[cbp + SXH + 0] - x[cbp + 0])
                      + fabsf(x[cbp + SXH + 1] - x[cbp + 1])
                      + fabsf(x[cbp + SXH + 2] - x[cbp + 2]);
        const float w = expf(-50.0f * g);
        const int pb = (b * Hh + y) * Ww + xi;
        val = w * fabsf(pred[pb + Ww] - pred[pb]);
    }
    __shared__ float red[256];
    red[threadIdx.x] = val;
    __syncthreads();
    #pragma unroll
    for (int off = 128; off > 0; off >>= 1) {
        if (threadIdx.x < off) red[threadIdx.x] += red[threadIdx.x + off];
        __syncthreads();
    }
    if (threadIdx.x == 0) partials[blockIdx.x] = red[0];
}

// Single-wave final reduction. Fixed strided per-lane summation (deterministic),
// then the 32 lane partials are collapsed with V_WMMA_F32_16X16X4_F32:
// A(16x4) holds the partials (lanes 0-15 -> K0, lanes 16-31 -> K2, K1/K3 = 0),
// B = all-ones  =>  sum over all rows of D = sum of ALL A elements (layout-proof).
// D layout: VGPR r @ lane 0 = D[r][0] (M=0..7), @ lane 16 = D[8+r][0] (M=8..15).
__global__ __launch_bounds__(32) void final_kernel(
    const float* __restrict__ clp, int ncl,
    const float* __restrict__ gp,  int ng,
    const int* __restrict__ epoch_p, float* __restrict__ out)
{
    const int lane = threadIdx.x;
    float pcl = 0.0f, pg = 0.0f;
    for (int i = lane; i < ncl; i += 32) pcl += clp[i];
    for (int i = lane; i < ng;  i += 32) pg  += gp[i];

    v2f ones; ones[0] = 1.0f; ones[1] = 1.0f;
    v8f cz = {};
    v2f a;  a[0] = pcl; a[1] = 0.0f;
    v8f d1 = __builtin_amdgcn_wmma_f32_16x16x4_f32(false, a, false, ones,
                                                   (short)0, cz, false, false);
    a[0] = pg;
    v8f d2 = __builtin_amdgcn_wmma_f32_16x16x4_f32(false, a, false, ones,
                                                   (short)0, cz, false, false);
    float s1 = 0.0f, s2 = 0.0f;
    #pragma unroll
    for (int r = 0; r < 8; ++r) { s1 += d1[r]; s2 += d2[r]; }

    __shared__ float red[64];
    red[lane]      = s1;
    red[32 + lane] = s2;
    __syncthreads();
    if (lane == 0) {
        const float total_cl = red[0] + red[16];        // M=0..7 + M=8..15 at N=0
        const float total_g  = red[32] + red[48];
        const int   K = dev_K(epoch_p[0]);
        // mean(cl) over (2,3,49,496,496) with NV/K scale
        const float mean_cl = total_cl * (49.0f / (float)K) / 72328704.0f;
        // (lx+ly)/2 with equal counts 2*496*495 each
        const float gl = total_g / 982080.0f;
        out[0] = mean_cl + 0.1f * gl;
    }
}

extern "C" void kernel_launch(void* const* d_in, const int* in_sizes, int n_in,
                              void* d_out, int out_size, void* d_ws, size_t ws_size,
                              hipStream_t stream)
{
    const float* pred  = (const float*)d_in[0];
    const float* x     = (const float*)d_in[1];
    const int*   epoch = (const int*)d_in[2];

    float* ws      = (float*)d_ws;
    float* cl_part = ws;                       // 7688 floats
    float* g_part  = ws + 8192;                // 3837 floats

    const int npix    = 2 * HC * WC;           // 492032
    const int nblk_cl = npix / 64;             // 7688 (exact)
    const int ntot_g  = 2 * 496 * 495 * 2;     // 982080
    const int nblk_g  = (ntot_g + 255) / 256;  // 3837

    color_kernel<<<nblk_cl, 64, 0, stream>>>(pred, x, epoch, cl_part);
    grad_kernel <<<nblk_g, 256, 0, stream>>>(pred, x, g_part);
    final_kernel<<<1, 32, 0, stream>>>(cl_part, nblk_cl, g_part, nblk_g, epoch,
                                       (float*)d_out);
}